// DecoderRNN_51694226375471
// MI455X (gfx1250) — compile-verified
//
#include <hip/hip_runtime.h>
#include <math.h>

// ---- problem dims (fixed by reference) ----
#define BB 64
#define SS 128
#define EE 512
#define HH 1024
#define VV 50257
#define G4H 4096

typedef __attribute__((ext_vector_type(2))) float v2f;
typedef __attribute__((ext_vector_type(8))) float v8f;

__device__ __forceinline__ v2f ld2(const float* p) { return *(const v2f*)p; }
// streaming (single-use) loads: TH=NT so they don't evict L2-resident reuse data
__device__ __forceinline__ v2f ld2_nt(const float* p) {
  return __builtin_nontemporal_load((const v2f*)p);
}

__device__ __forceinline__ v8f wmma_f32(v2f a, v2f b, v8f c) {
  // D = A(16x4 f32) * B(4x16 f32) + C(16x16 f32)
  return __builtin_amdgcn_wmma_f32_16x16x4_f32(false, a, false, b, (short)0, c,
                                               false, false);
}

// ---------------------------------------------------------------------------
// Kernel 1: hW[b,n] = sum_k h0[b,k]*attn_W[n,k] + attn_Wb[n]   (64x1024,K=1024)
// 64 waves: mi in 0..3 (M tile), nc in 0..15 (64-col chunk)
// ---------------------------------------------------------------------------
__global__ void k_hW(const float* __restrict__ h0, const float* __restrict__ Wa,
                     const float* __restrict__ ba, float* __restrict__ hW) {
  int w    = blockIdx.x * 8 + (threadIdx.x >> 5);
  int lane = threadIdx.x & 31;
  int half = lane >> 4, l15 = lane & 15;
  int mi = w & 3, nc = w >> 2;
  int n0 = nc * 64;
  const float* ap = h0 + (mi * 16 + l15) * HH + 2 * half;
  v8f acc[4] = {};
  for (int k0 = 0; k0 < HH; k0 += 4) {
    v2f a = ld2(ap + k0);
#pragma unroll
    for (int t = 0; t < 4; ++t)
      acc[t] = wmma_f32(a, ld2(Wa + (n0 + 16 * t + l15) * HH + 2 * half + k0), acc[t]);
  }
#pragma unroll
  for (int t = 0; t < 4; ++t) {
    int col = n0 + 16 * t + l15;
    float bias = ba[col];
#pragma unroll
    for (int r = 0; r < 8; ++r) {
      int row = mi * 16 + r + 8 * half;
      hW[row * HH + col] = acc[t][r] + bias;
    }
  }
}

// ---------------------------------------------------------------------------
// Kernel 2: fused energy GEMM + tanh + v-dot.
// rows r = s*B+b of enc (8192 rows, K=1024); per wave: 16 rows x all 1024 cols.
// score[b,s] = sum_n v[n]*tanh(enc[r]·U[n] + Ub[n] + hW[b,n])
// 512 waves => 64 blocks x 8 waves. attn_U (4MB) stays L2-resident across waves.
// ---------------------------------------------------------------------------
__global__ void k_energy_score(const float* __restrict__ enc,
                               const float* __restrict__ U,
                               const float* __restrict__ Ub,
                               const float* __restrict__ hW,
                               const float* __restrict__ vvec,
                               float* __restrict__ score) {
  int w    = blockIdx.x * 8 + (threadIdx.x >> 5);
  int lane = threadIdx.x & 31;
  int half = lane >> 4, l15 = lane & 15;
  int r0 = w * 16;
  const float* ap = enc + (r0 + l15) * HH + 2 * half;
  float sc[8] = {};
  for (int nc = 0; nc < 16; ++nc) {
    int n0 = nc * 64;
    v8f acc[4] = {};
    for (int k0 = 0; k0 < HH; k0 += 4) {
      v2f a = ld2(ap + k0);
#pragma unroll
      for (int t = 0; t < 4; ++t)
        acc[t] = wmma_f32(a, ld2(U + (n0 + 16 * t + l15) * HH + 2 * half + k0), acc[t]);
    }
#pragma unroll
    for (int t = 0; t < 4; ++t) {
      int col = n0 + 16 * t + l15;
      float ub = Ub[col], vv = vvec[col];
#pragma unroll
      for (int r = 0; r < 8; ++r) {
        int rr = r0 + r + 8 * half;     // global row = s*B+b
        int b  = rr & (BB - 1);
        float e = tanhf(acc[t][r] + ub + hW[b * HH + col]);
        sc[r] += vv * e;
      }
    }
  }
  // reduce over the 16 lanes of each half (columns), rows live per-register
#pragma unroll
  for (int r = 0; r < 8; ++r) {
    float x = sc[r];
    x += __shfl_xor(x, 1, 32);
    x += __shfl_xor(x, 2, 32);
    x += __shfl_xor(x, 4, 32);
    x += __shfl_xor(x, 8, 32);
    sc[r] = x;
  }
  if (l15 == 0) {
#pragma unroll
    for (int r = 0; r < 8; ++r) {
      int rr = r0 + r + 8 * half;
      int s  = rr >> 6;         // / BB
      int b  = rr & (BB - 1);
      score[b * SS + s] = sc[r];
    }
  }
}

// ---------------------------------------------------------------------------
// Kernel 3: per-b softmax over S=128 + context[b,h] = sum_s w*enc[s,b,h]
// 64 blocks x 256 threads
// ---------------------------------------------------------------------------
__global__ void k_softmax_ctx(const float* __restrict__ score,
                              const float* __restrict__ enc,
                              float* __restrict__ attn_w,
                              float* __restrict__ ctx) {
  __shared__ float wsm[SS];
  __shared__ float red[16];
  int b = blockIdx.x, tid = threadIdx.x;
  float val = (tid < SS) ? score[b * SS + tid] : -INFINITY;
  float m = val;
  for (int off = 16; off; off >>= 1) m = fmaxf(m, __shfl_xor(m, off, 32));
  if ((tid & 31) == 0) red[tid >> 5] = m;
  __syncthreads();
  if (tid == 0) {
    float mm = red[0];
    for (int i = 1; i < 8; ++i) mm = fmaxf(mm, red[i]);
    red[0] = mm;
  }
  __syncthreads();
  float mx = red[0];
  float e = (tid < SS) ? expf(val - mx) : 0.0f;
  float s = e;
  for (int off = 16; off; off >>= 1) s += __shfl_xor(s, off, 32);
  if ((tid & 31) == 0) red[8 + (tid >> 5)] = s;
  __syncthreads();
  if (tid == 0) {
    float ss = 0.f;
    for (int i = 0; i < 8; ++i) ss += red[8 + i];
    red[8] = ss;
  }
  __syncthreads();
  float inv = 1.0f / red[8];
  if (tid < SS) {
    float wv = e * inv;
    wsm[tid] = wv;
    attn_w[b * SS + tid] = wv;   // [B,1,S]
  }
  __syncthreads();
  for (int h = tid; h < HH; h += 256) {
    float acc = 0.f;
#pragma unroll 4
    for (int s0 = 0; s0 < SS; ++s0)
      acc += wsm[s0] * enc[(s0 * BB + b) * HH + h];
    ctx[b * HH + h] = acc;
  }
}

// ---------------------------------------------------------------------------
// Kernel 4: gates = [emb;ctx]·W_ihT + h·W_hhT + b_ih + b_hh (64x4096, K=2560)
// concat never materialized: uniform branch on k0 picks A source / weight.
// 256 waves => 32 blocks x 8 waves. Weights are single-use -> NT loads.
// ---------------------------------------------------------------------------
__global__ void k_gates(const int* __restrict__ seq, const float* __restrict__ emb,
                        const float* __restrict__ ctx, const float* __restrict__ h0,
                        const float* __restrict__ Wih, const float* __restrict__ Whh,
                        const float* __restrict__ bih, const float* __restrict__ bhh,
                        float* __restrict__ gates) {
  int w    = blockIdx.x * 8 + (threadIdx.x >> 5);
  int lane = threadIdx.x & 31;
  int half = lane >> 4, l15 = lane & 15;
  int mi = w & 3, nc = w >> 2;       // nc 0..63
  int n0 = nc * 64;
  int m  = mi * 16 + l15;
  int erow = seq[m];
  const float* ea = emb + erow * EE + 2 * half;          // k in [0,512)
  const float* ca = ctx + m * HH - EE + 2 * half;        // k in [512,1536)
  const float* ha = h0 + m * HH - (EE + HH) + 2 * half;  // k in [1536,2560)
  v8f acc[4] = {};
  for (int k0 = 0; k0 < EE + HH + HH; k0 += 4) {
    v2f a;
    if (k0 < EE)            a = ld2(ea + k0);
    else if (k0 < EE + HH)  a = ld2(ca + k0);
    else                    a = ld2(ha + k0);
    if (k0 < EE + HH) {
#pragma unroll
      for (int t = 0; t < 4; ++t)
        acc[t] = wmma_f32(a, ld2_nt(Wih + (n0 + 16 * t + l15) * (EE + HH) + 2 * half + k0), acc[t]);
    } else {
#pragma unroll
      for (int t = 0; t < 4; ++t)
        acc[t] = wmma_f32(a, ld2_nt(Whh + (n0 + 16 * t + l15) * HH + 2 * half + (k0 - EE - HH)), acc[t]);
    }
  }
#pragma unroll
  for (int t = 0; t < 4; ++t) {
    int col = n0 + 16 * t + l15;
    float bias = bih[col] + bhh[col];
#pragma unroll
    for (int r = 0; r < 8; ++r) {
      int row = mi * 16 + r + 8 * half;
      gates[row * G4H + col] = acc[t][r] + bias;
    }
  }
}

// ---------------------------------------------------------------------------
// Kernel 5: LSTM pointwise
// ---------------------------------------------------------------------------
__global__ void k_lstm(const float* __restrict__ gates, const float* __restrict__ c0,
                       float* __restrict__ h_new, float* __restrict__ c_new) {
  int idx = blockIdx.x * blockDim.x + threadIdx.x;
  if (idx >= BB * HH) return;
  int b = idx >> 10, hh = idx & (HH - 1);
  const float* g = gates + b * G4H;
  float gi = g[hh], gf = g[HH + hh], gg = g[2 * HH + hh], go = g[3 * HH + hh];
  float si = 1.f / (1.f + expf(-gi));
  float sf = 1.f / (1.f + expf(-gf));
  float so = 1.f / (1.f + expf(-go));
  float cn = sf * c0[idx] + si * tanhf(gg);
  float hn = so * tanhf(cn);
  c_new[idx] = cn;
  h_new[idx] = hn;
}

// ---------------------------------------------------------------------------
// Kernel 6: logits = h_new·out_WT + out_b  (64x50257, K=1024) — HBM-bound.
// out_W (206MB, single use) streamed with NT loads so the 192MB L2 keeps the
// reused tensors (logits, enc, attn_U) instead. 3144 waves => 393 blocks.
// ---------------------------------------------------------------------------
__global__ void k_logits(const float* __restrict__ hn, const float* __restrict__ Wo,
                         const float* __restrict__ bo, float* __restrict__ logits) {
  int w    = blockIdx.x * 8 + (threadIdx.x >> 5);
  int lane = threadIdx.x & 31;
  int half = lane >> 4, l15 = lane & 15;
  int mi = w & 3, nc = w >> 2;     // nc 0..785
  int n0 = nc * 64;
  const float* ap = hn + (mi * 16 + l15) * HH + 2 * half;
  int  colv[4];
  bool ok[4];
  const float* bp[4];
#pragma unroll
  for (int t = 0; t < 4; ++t) {
    colv[t] = n0 + 16 * t + l15;
    ok[t]   = colv[t] < VV;
    bp[t]   = Wo + (ok[t] ? colv[t] : 0) * HH + 2 * half;
  }
  v8f acc[4] = {};
  for (int k0 = 0; k0 < HH; k0 += 4) {
    v2f a = ld2(ap + k0);
#pragma unroll
    for (int t = 0; t < 4; ++t) {
      v2f bb = ld2_nt(bp[t] + k0);
      acc[t] = wmma_f32(a, bb, acc[t]);
    }
  }
#pragma unroll
  for (int t = 0; t < 4; ++t) {
    if (!ok[t]) continue;
    float bias = bo[colv[t]];
#pragma unroll
    for (int r = 0; r < 8; ++r) {
      int row = mi * 16 + r + 8 * half;
      logits[row * VV + colv[t]] = acc[t][r] + bias;
    }
  }
}

// ---------------------------------------------------------------------------
// Kernel 7: log_softmax per row (64 blocks x 1024 threads); logits rows are
// L2-resident (12.9MB) thanks to NT policy on out_W.
// ---------------------------------------------------------------------------
__global__ void k_logsoftmax(const float* __restrict__ logits, float* __restrict__ out) {
  __shared__ float red[32];
  int b = blockIdx.x, tid = threadIdx.x;
  const float* row = logits + (size_t)b * VV;
  float m = -INFINITY;
  for (int v = tid; v < VV; v += 1024) m = fmaxf(m, row[v]);
  for (int off = 16; off; off >>= 1) m = fmaxf(m, __shfl_xor(m, off, 32));
  if ((tid & 31) == 0) red[tid >> 5] = m;
  __syncthreads();
  if (tid < 32) {
    float mm = red[tid];
    for (int off = 16; off; off >>= 1) mm = fmaxf(mm, __shfl_xor(mm, off, 32));
    if (tid == 0) red[0] = mm;
  }
  __syncthreads();
  float mx = red[0];
  __syncthreads();
  float s = 0.f;
  for (int v = tid; v < VV; v += 1024) s += expf(row[v] - mx);
  for (int off = 16; off; off >>= 1) s += __shfl_xor(s, off, 32);
  if ((tid & 31) == 0) red[tid >> 5] = s;
  __syncthreads();
  if (tid < 32) {
    float ss = red[tid];
    for (int off = 16; off; off >>= 1) ss += __shfl_xor(ss, off, 32);
    if (tid == 0) red[0] = ss;
  }
  __syncthreads();
  float lse = mx + logf(red[0]);
  float* orow = out + (size_t)b * VV;
  for (int v = tid; v < VV; v += 1024) orow[v] = row[v] - lse;
}

// ---------------------------------------------------------------------------
extern "C" void kernel_launch(void* const* d_in, const int* in_sizes, int n_in,
                              void* d_out, int out_size, void* d_ws, size_t ws_size,
                              hipStream_t stream) {
  const int*   input_seq = (const int*)d_in[0];
  const float* h0        = (const float*)d_in[1];
  const float* c0        = (const float*)d_in[2];
  const float* enc       = (const float*)d_in[3];
  const float* emb       = (const float*)d_in[4];
  const float* W_ih      = (const float*)d_in[5];
  const float* W_hh      = (const float*)d_in[6];
  const float* b_ih      = (const float*)d_in[7];
  const float* b_hh      = (const float*)d_in[8];
  const float* attn_W    = (const float*)d_in[9];
  const float* attn_Wb   = (const float*)d_in[10];
  const float* attn_U    = (const float*)d_in[11];
  const float* attn_Ub   = (const float*)d_in[12];
  const float* attn_v    = (const float*)d_in[13];
  const float* out_W     = (const float*)d_in[14];
  const float* out_b     = (const float*)d_in[15];

  // output layout: [output(B*V) | h_new(B*H) | c_new(B*H) | attn_w(B*S)]
  float* out_log = (float*)d_out;
  float* h_new   = out_log + (size_t)BB * VV;
  float* c_new   = h_new + BB * HH;
  float* attn_w  = c_new + BB * HH;

  // workspace layout (floats)
  float* wsf      = (float*)d_ws;
  float* ws_hW    = wsf;                      // 64*1024
  float* ws_score = ws_hW + BB * HH;          // 64*128
  float* ws_ctx   = ws_score + BB * SS;       // 64*1024
  float* ws_gates = ws_ctx + BB * HH;         // 64*4096
  float* ws_logit = ws_gates + BB * G4H;      // 64*50257

  k_hW<<<8, 256, 0, stream>>>(h0, attn_W, attn_Wb, ws_hW);
  k_energy_score<<<64, 256, 0, stream>>>(enc, attn_U, attn_Ub, ws_hW, attn_v, ws_score);
  k_softmax_ctx<<<64, 256, 0, stream>>>(ws_score, enc, attn_w, ws_ctx);
  k_gates<<<32, 256, 0, stream>>>(input_seq, emb, ws_ctx, h0, W_ih, W_hh, b_ih, b_hh, ws_gates);
  k_lstm<<<256, 256, 0, stream>>>(ws_gates, c0, h_new, c_new);
  k_logits<<<393, 256, 0, stream>>>(h_new, out_W, out_b, ws_logit);
  k_logsoftmax<<<64, 1024, 0, stream>>>(ws_logit, out_log);
}